// Attention_45260365365743
// MI455X (gfx1250) — compile-verified
//
#include <hip/hip_runtime.h>
#include <math.h>

typedef __attribute__((ext_vector_type(2))) float v2f;
typedef __attribute__((ext_vector_type(4))) float v4f;
typedef __attribute__((ext_vector_type(8))) float v8f;

// Async copy of 16 bytes/lane from global to LDS (tracked with ASYNCcnt).
__device__ __forceinline__ void async_g2lds_b128(unsigned ldsOff, const float* g) {
    asm volatile("global_load_async_to_lds_b128 %0, %1, off"
                 :: "v"(ldsOff), "v"(g) : "memory");
}
__device__ __forceinline__ void wait_asynccnt0() {
    asm volatile("s_wait_asynccnt 0" ::: "memory");
}

// ---------------------------------------------------------------------------
// Pass A: fcb[b,h] = fc[h] + type_embed[b,h]; also zero d_out (same BH size).
// ---------------------------------------------------------------------------
__global__ void prep_kernel(const float* __restrict__ fc,
                            const float* __restrict__ te,
                            float* __restrict__ fcb,
                            float* __restrict__ out,
                            int BH, int H) {
    int i = blockIdx.x * blockDim.x + threadIdx.x;
    if (i < BH) {
        fcb[i] = fc[i % H] + te[i];
        out[i] = 0.0f;
    }
}

// ---------------------------------------------------------------------------
// Pass B: q[b,s] = hidden[b,s,:] . fcb[b,:], masked to -10000 where mask==0.
// One wave per 16-row s-tile, 8 waves (one batch) per block.
//   * fcb[b,:] staged in LDS once per block via global_load_async_to_lds_b128.
//   * A operand: one global_load_b128 per 8 k-values. Since WMMA accumulation
//     is order-free over k, each WMMA consumes a shuffle-free column set:
//       WMMA1: cols {k0, k0+1, k0+4, k0+5}   -> A = x.xy for all lanes
//       WMMA2: cols {k0+2, k0+3, k0+6, k0+7} -> A = x.zw for all lanes
//     (lane<16 loads cols k0..k0+3, lane>=16 loads k0+4..k0+7, which exactly
//     matches the ISA 16x4 f32 A layout for these column sets)
//   * B operand (fcb chunk broadcast across all 16 N columns) is a single
//     16B-aligned ds_load_b128 at fcbLDS + k0 + 4*kh per 8 k-values.
// Every column of the f32 accumulator equals q for these 16 rows.
// C/D layout: lane L, VGPR r holds M = 8*(L>>4)+r -> lanes 0,16 write 8 each.
// ---------------------------------------------------------------------------
__global__ void __launch_bounds__(256)
qk_kernel(const float* __restrict__ hidden,
          const int*   __restrict__ mask,
          const float* __restrict__ fcb,
          float*       __restrict__ q,
          int S, int H) {
    __shared__ __align__(16) float fcbLDS[1024];

    const int t    = threadIdx.x;
    const int lane = t & 31;
    const int wave = t >> 5;
    const int tilesPerBatch = S >> 4;             // 256: multiple of 8 -> one batch/block
    const int tile = blockIdx.x * 8 + wave;
    const int b    = tile / tilesPerBatch;
    const int srow = (tile - b * tilesPerBatch) << 4;
    const int m  = lane & 15;
    const int kh = lane >> 4;
    const int bBlock = (blockIdx.x * 8) / tilesPerBatch;

    // Stage fcb[bBlock, :] into LDS (async, ASYNCcnt-tracked).
    for (int i = t * 4; i < H; i += 1024) {
        async_g2lds_b128((unsigned)(size_t)&fcbLDS[i],
                         fcb + (size_t)bBlock * H + i);
    }
    wait_asynccnt0();
    __syncthreads();

    const float* rowBase = hidden + ((size_t)b * S + (size_t)(srow + m)) * H;

#if defined(__gfx1250__) && __has_builtin(__builtin_amdgcn_wmma_f32_16x16x4_f32)
    const float* Arow = rowBase + 4 * kh;   // lane covers cols k0+4*kh .. +3
    const float* Bv   = fcbLDS + 4 * kh;

    v8f acc = {};
    for (int k0 = 0; k0 < H; k0 += 8) {
        v4f x  = __builtin_nontemporal_load((const v4f*)(Arow + k0));
        v4f bv = *(const v4f*)(Bv + k0);    // ds_load_b128 (broadcast per half)
        v2f a1 = {x.x, x.y};
        v2f a2 = {x.z, x.w};
        v2f b1 = {bv.x, bv.y};
        v2f b2 = {bv.z, bv.w};
        acc = __builtin_amdgcn_wmma_f32_16x16x4_f32(
            false, a1, false, b1, (short)0, acc, false, false);
        acc = __builtin_amdgcn_wmma_f32_16x16x4_f32(
            false, a2, false, b2, (short)0, acc, false, false);
    }
    if (m == 0) {   // lanes 0 and 16: 8 rows each
        for (int r = 0; r < 8; ++r) {
            int s = srow + 8 * kh + r;
            float v = acc[r];
            if (mask[(size_t)b * S + s] == 0) v = -10000.0f;
            q[(size_t)b * S + s] = v;
        }
    }
#else
    // VALU fallback: lanes L and L+16 split k-space for row (L&15).
    const float* A   = rowBase + 2 * kh;
    const float* Bvf = fcbLDS + 2 * kh;
    float p = 0.0f;
    for (int k0 = 0; k0 < H; k0 += 4) {
        p += A[k0] * Bvf[k0] + A[k0 + 1] * Bvf[k0 + 1];
    }
    p += __shfl_xor(p, 16, 32);
    if (kh == 0) {
        int s = srow + m;
        if (mask[(size_t)b * S + s] == 0) p = -10000.0f;
        q[(size_t)b * S + s] = p;
    }
#endif
}

// ---------------------------------------------------------------------------
// Pass C: softmax over S per batch. One block (256 threads) per batch;
// each thread owns S/256 strided elements; LDS tree reductions.
// ---------------------------------------------------------------------------
__global__ void __launch_bounds__(256)
softmax_kernel(float* __restrict__ q, int S) {
    const int b = blockIdx.x;
    const int t = threadIdx.x;
    float* qb = q + (size_t)b * S;
    const int per = S / 256;           // 16 for S=4096
    float vals[16];

    float mx = -INFINITY;
    for (int i = 0; i < per; ++i) {
        float v = qb[t + i * 256];
        vals[i] = v;
        mx = fmaxf(mx, v);
    }
    __shared__ float red[256];
    red[t] = mx;
    __syncthreads();
    for (int off = 128; off > 0; off >>= 1) {
        if (t < off) red[t] = fmaxf(red[t], red[t + off]);
        __syncthreads();
    }
    mx = red[0];
    __syncthreads();

    float sum = 0.0f;
    for (int i = 0; i < per; ++i) {
        float e = expf(vals[i] - mx);
        vals[i] = e;
        sum += e;
    }
    red[t] = sum;
    __syncthreads();
    for (int off = 128; off > 0; off >>= 1) {
        if (t < off) red[t] += red[t + off];
        __syncthreads();
    }
    float inv = 1.0f / red[0];
    for (int i = 0; i < per; ++i) qb[t + i * 256] = vals[i] * inv;
}

// ---------------------------------------------------------------------------
// Pass D: h[b,:] += sum_{s in chunk} w[b,s] * hidden[b,s,:].
// Block = 256 threads * float4 = 1024 columns (all of H); each block owns a
// 128-row s-chunk; w staged in LDS via async copy (one ds_load_b128 per
// 4 rows); every s-iteration reads one fully-coalesced 4KB row
// (nontemporal: single-touch stream). Combined with global_atomic_add_f32.
// ---------------------------------------------------------------------------
__global__ void __launch_bounds__(256)
hv_kernel(const float* __restrict__ hidden,
          const float* __restrict__ w,
          float*       __restrict__ out,
          int S, int H, int chunk) {
    __shared__ __align__(16) float wLDS[128];

    const int b  = blockIdx.y;
    const int s0 = blockIdx.x * chunk;
    const int t  = threadIdx.x;

    for (int i = t * 4; i < chunk; i += 1024) {
        async_g2lds_b128((unsigned)(size_t)&wLDS[i],
                         w + (size_t)b * S + s0 + i);
    }
    wait_asynccnt0();
    __syncthreads();

    const int h = t * 4;
    if (h < H) {
        const float* base = hidden + ((size_t)b * S + s0) * H + h;
        v4f acc = {};
        for (int s = 0; s < chunk; s += 4) {
            v4f wv = *(const v4f*)&wLDS[s];
            acc += __builtin_nontemporal_load((const v4f*)(base + (size_t)(s + 0) * H)) * wv.x;
            acc += __builtin_nontemporal_load((const v4f*)(base + (size_t)(s + 1) * H)) * wv.y;
            acc += __builtin_nontemporal_load((const v4f*)(base + (size_t)(s + 2) * H)) * wv.z;
            acc += __builtin_nontemporal_load((const v4f*)(base + (size_t)(s + 3) * H)) * wv.w;
        }
        float* o = out + (size_t)b * H + h;
        atomicAdd(o + 0, acc.x);
        atomicAdd(o + 1, acc.y);
        atomicAdd(o + 2, acc.z);
        atomicAdd(o + 3, acc.w);
    }
}

extern "C" void kernel_launch(void* const* d_in, const int* in_sizes, int n_in,
                              void* d_out, int out_size, void* d_ws, size_t ws_size,
                              hipStream_t stream) {
    const float* hidden = (const float*)d_in[0];
    const int*   mask   = (const int*)d_in[1];
    const float* te     = (const float*)d_in[2];
    const float* fc     = (const float*)d_in[3];

    const int H  = in_sizes[3];          // fc: [H,1]
    const int BH = in_sizes[2];          // type_embed: [B,H,1]
    const int B  = BH / H;
    const int S  = in_sizes[1] / B;      // mask: [B,S]

    float* out = (float*)d_out;          // [B,H]
    float* fcb = (float*)d_ws;           // [B,H]
    float* q   = fcb + BH;               // [B,S] (q, then overwritten by w)

    // Pass A: fcb + zero output
    prep_kernel<<<(BH + 255) / 256, 256, 0, stream>>>(fc, te, fcb, out, BH, H);

    // Pass B: masked logits via WMMA (f32 16x16x4), fcb staged in LDS
    int tiles = B * (S / 16);            // 8192
    qk_kernel<<<tiles / 8, 256, 0, stream>>>(hidden, mask, fcb, q, S, H);

    // Pass C: softmax per batch
    softmax_kernel<<<B, 256, 0, stream>>>(q, S);

    // Pass D: weighted sum back over S
    const int chunk = 128;
    dim3 gridD(S / chunk, B);
    hv_kernel<<<gridD, 256, 0, stream>>>(hidden, q, out, S, H, chunk);
}